// BayesianLoss_498216206855
// MI455X (gfx1250) — compile-verified
//
#include <hip/hip_runtime.h>

typedef float v8f __attribute__((ext_vector_type(8)));
typedef float v2f __attribute__((ext_vector_type(2)));

#define HH 384
#define WW 384
#define NPTS 1024
#define NCHUNK (NPTS / 16)                 // 64 chunks of 16 points
#define PIX_PER_BLOCK 128
#define NBLOCKS ((HH * WW) / PIX_PER_BLOCK)   // 1152
#define LOG2E 1.4426950408889634f

// ---------------------------------------------------------------------------
// zero the workspace accumulators (expected_counts[1024] + bg scalar)
// ---------------------------------------------------------------------------
__global__ void bayes_zero_ws(float* ws) {
    int i = blockIdx.x * blockDim.x + threadIdx.x;
    if (i < NPTS + 1) ws[i] = 0.0f;
}

// ---------------------------------------------------------------------------
// main kernel: per block of 128 pixels, two passes over all 1024 points.
// dist^2 via V_WMMA_F32_16X16X4_F32:
//   A row m (pixel): [gx, gy, gx^2+gy^2, 1]
//   B col n (point): [-2x, -2y, 1, x^2+y^2]
//   D[m][n] = (gx-x)^2 + (gy-y)^2
// Wave32 layouts (ISA 7.12.2):
//   A 16x4 f32: lanes 0-15 hold (K0,K1)=(gx,gy) of row=lane;
//               lanes 16-31 hold (K2,K3)=(|g|^2,1) of row=lane-16
//   B 4x16 f32: lanes 0-15 hold (K0,K1)=(-2x,-2y) of col=lane;
//               lanes 16-31 hold (K2,K3)=(1,|p|^2) of col=lane-16
//   D 16x16 f32: VGPR v: lanes 0-15 -> M=v, lanes 16-31 -> M=v+8; N=lane%16
//
// B is pre-packed in LDS in lane-native order: Bpack[chunk*32 + lane] is the
// exact float2 that lane needs => one unconditional ds_load_b64 per chunk,
// no exec-mask divergence in the hot loop, all 64 LDS banks hit.
// ---------------------------------------------------------------------------
__launch_bounds__(256)
__global__ void bayes_main(const float* __restrict__ pred,
                           const float* __restrict__ pts,
                           float* __restrict__ ec,      // [NPTS] global accum
                           float* __restrict__ bgAcc)   // [1] global accum
{
    __shared__ v2f   Bpack[NCHUNK * 32];          // 16 KB
    __shared__ float contrib[NPTS];               // 4 KB
    __shared__ float likS[PIX_PER_BLOCK];
    __shared__ float minD[PIX_PER_BLOCK];
    __shared__ float wgt[PIX_PER_BLOCK];
    __shared__ float bgLds;

    const int tid   = threadIdx.x;
    const int wv_id = tid >> 5;       // wave 0..7
    const int lane  = tid & 31;
    const int half  = lane >> 4;      // 0: lanes 0-15, 1: lanes 16-31
    const int m     = lane & 15;      // row (A) / col (B) index inside tile

    // ---- stage points into LDS, pre-packed per-lane for the B matrix ----
    for (int n = tid; n < NPTS; n += 256) {
        float x = pts[2 * n + 0];
        float y = pts[2 * n + 1];
        int   c = n >> 4;             // chunk
        int col = n & 15;             // column within chunk
        v2f lo; lo.x = -2.0f * x; lo.y = -2.0f * y;
        v2f hi; hi.x = 1.0f;      hi.y = x * x + y * y;
        Bpack[c * 32 + col]      = lo;   // consumed by lanes 0-15
        Bpack[c * 32 + 16 + col] = hi;   // consumed by lanes 16-31
        contrib[n] = 0.0f;
    }
    if (tid == 0) bgLds = 0.0f;

    // ---- A matrix (constant across the whole kernel for this lane) ----
    const int pixRow = blockIdx.x * PIX_PER_BLOCK + wv_id * 16 + m;
    const float gy = (float)(pixRow / WW);
    const float gx = (float)(pixRow % WW);
    v2f A;
    if (half == 0) { A.x = gx;                A.y = gy;   }
    else           { A.x = gx * gx + gy * gy; A.y = 1.0f; }

    __syncthreads();

    const float c1 = -LOG2E / 128.0f;   // exp(-d2/128) == exp2(c1*d2)

    // ================= pass 1: lik_sum & min dist per pixel =================
    float acc[8], mn[8];
    #pragma unroll
    for (int v = 0; v < 8; ++v) { acc[v] = 0.0f; mn[v] = 3.4e38f; }

    {
        v2f Bcur = Bpack[lane];                       // prefetch chunk 0
        for (int nc = 0; nc < NCHUNK; ++nc) {
            v2f Bnext = Bpack[((nc + 1) & (NCHUNK - 1)) * 32 + lane];
            v8f C = {};
            v8f D = __builtin_amdgcn_wmma_f32_16x16x4_f32(
                        false, A, false, Bcur, (short)0, C, false, false);
            #pragma unroll
            for (int v = 0; v < 8; ++v) {
                float d2 = D[v];
                acc[v] += __builtin_amdgcn_exp2f(d2 * c1);
                mn[v]   = fminf(mn[v], d2);
            }
            Bcur = Bnext;
        }
    }

    // reduce across the 16-lane halves (sum over point columns / min)
    #pragma unroll
    for (int v = 0; v < 8; ++v) {
        #pragma unroll
        for (int mask = 1; mask < 16; mask <<= 1) {
            acc[v] += __shfl_xor(acc[v], mask, 32);
            mn[v]   = fminf(mn[v], __shfl_xor(mn[v], mask, 32));
        }
    }
    if (m == 0) {   // lane 0 writes rows 0..7, lane 16 writes rows 8..15
        #pragma unroll
        for (int v = 0; v < 8; ++v) {
            likS[wv_id * 16 + half * 8 + v] = acc[v];
            minD[wv_id * 16 + half * 8 + v] = mn[v];
        }
    }
    __syncthreads();

    // ---- per-pixel: posterior weight + background term ----
    if (tid < PIX_PER_BLOCK) {
        const int p = blockIdx.x * PIX_PER_BLOCK + tid;
        float ls = fmaxf(likS[tid], 1e-8f);
        float pr = pred[p];
        wgt[tid] = pr / ls;
        float md = sqrtf(minD[tid]);
        float t  = md - 76.8f;
        float bgl = __builtin_amdgcn_exp2f(t * t * c1);
        float bgp = bgl / (ls + bgl);
        atomicAdd(&bgLds, bgp * pr);
    }
    __syncthreads();

    // preload this lane's 8 row weights (constant over the n-loop)
    float wrow[8];
    #pragma unroll
    for (int v = 0; v < 8; ++v) wrow[v] = wgt[wv_id * 16 + half * 8 + v];

    // ============ pass 2: expected_counts[n] += lik * pred/lik_sum ===========
    {
        v2f Bcur = Bpack[lane];                       // prefetch chunk 0
        for (int nc = 0; nc < NCHUNK; ++nc) {
            v2f Bnext = Bpack[((nc + 1) & (NCHUNK - 1)) * 32 + lane];
            v8f C = {};
            v8f D = __builtin_amdgcn_wmma_f32_16x16x4_f32(
                        false, A, false, Bcur, (short)0, C, false, false);
            float partial = 0.0f;
            #pragma unroll
            for (int v = 0; v < 8; ++v)
                partial += __builtin_amdgcn_exp2f(D[v] * c1) * wrow[v];
            // combine the two halves holding the same column
            partial += __shfl_xor(partial, 16, 32);
            if (half == 0) atomicAdd(&contrib[nc * 16 + m], partial); // ds_add_f32
            Bcur = Bnext;
        }
    }
    __syncthreads();

    // ---- flush block-local accumulators to global ----
    for (int n = tid; n < NPTS; n += 256) atomicAdd(&ec[n], contrib[n]);
    if (tid == 0) atomicAdd(bgAcc, bgLds);
}

// ---------------------------------------------------------------------------
// finalize: loss = sum_n |ec[n] - 1| + |bg|
// ---------------------------------------------------------------------------
__launch_bounds__(1024)
__global__ void bayes_final(const float* __restrict__ ws, float* __restrict__ out) {
    __shared__ float red[32];
    const int t = threadIdx.x;
    float v = fabsf(ws[t] - 1.0f);
    #pragma unroll
    for (int mask = 1; mask < 32; mask <<= 1) v += __shfl_xor(v, mask, 32);
    if ((t & 31) == 0) red[t >> 5] = v;
    __syncthreads();
    if (t < 32) {
        float s = red[t];
        #pragma unroll
        for (int mask = 1; mask < 32; mask <<= 1) s += __shfl_xor(s, mask, 32);
        if (t == 0) out[0] = s + fabsf(ws[NPTS]);
    }
}

// ---------------------------------------------------------------------------
extern "C" void kernel_launch(void* const* d_in, const int* in_sizes, int n_in,
                              void* d_out, int out_size, void* d_ws, size_t ws_size,
                              hipStream_t stream) {
    const float* pred = (const float*)d_in[0];   // [1,1,384,384] row-major
    const float* pts  = (const float*)d_in[1];   // [1024,2] = (x,y)
    float* ws  = (float*)d_ws;                   // [0..1023]=ec, [1024]=bg
    float* out = (float*)d_out;

    bayes_zero_ws<<<(NPTS + 1 + 255) / 256, 256, 0, stream>>>(ws);
    bayes_main<<<NBLOCKS, 256, 0, stream>>>(pred, pts, ws, ws + NPTS);
    bayes_final<<<1, 1024, 0, stream>>>(ws, out);
}